// FilterBank_38345468018998
// MI455X (gfx1250) — compile-verified
//
#include <hip/hip_runtime.h>

typedef __attribute__((ext_vector_type(2))) float v2f;
typedef __attribute__((ext_vector_type(4))) float v4f;
typedef __attribute__((ext_vector_type(8))) float v8f;

#define T_LEN 2000
#define NB 9
#define NK 9
#define TILES 125              // 2000 / 16
#define WAVES_PER_BLOCK 4
#define XSTRIDE 2048           // 16-float zero lead-in + 2000 samples, padded
#define FSTRIDE 20             // fir-tile row stride (80B -> 16B aligned v4f rows)

__global__ __launch_bounds__(128)
void FilterBank_wmma_kernel(const float* __restrict__ x,
                            const float* __restrict__ a,
                            const float* __restrict__ b,
                            float* __restrict__ out, int nseq) {
    __shared__ float xsh[WAVES_PER_BLOCK][XSTRIDE];
    __shared__ float fsh[WAVES_PER_BLOCK][16 * FSTRIDE];

    const int tid  = threadIdx.x;
    const int wave = tid >> 5;
    const int lane = tid & 31;
    const int seq  = blockIdx.x * WAVES_PER_BLOCK + wave;
    if (seq >= nseq) return;              // uniform per wave; grid sized exactly

    float* xw = xsh[wave];
    float* fw = fsh[wave];

    // ---- stage x into LDS: 16-float zero lead-in + 2000 samples (b128) ----
    if (lane < 16) xw[lane] = 0.0f;
    const v4f* xg = (const v4f*)(x + (size_t)seq * T_LEN);
    for (int i = lane; i < T_LEN / 4; i += 32) {
        v4f v = xg[i];
        *(v4f*)&xw[16 + 4 * i] = v;
    }

    // ---- WMMA A fragments: bn[m][j] = b[m][j]/a[m][0], 3 chunks of 4 taps ----
    // A 16x4 f32 layout: lanes 0-15 hold M, VGPR0/VGPR1 = two K rows per half-wave.
    // K-row mapping swapped (tap j0+1 in .x, tap j0 in .y) so the B fragment can
    // be consumed directly from an ascending-address LDS pair load (no shuffles).
    const int   half = lane >> 4;
    const int   m    = lane & 15;
    const int   mclp = (m < NB) ? m : 0;
    const float ra0  = 1.0f / a[mclp * NK];
    v2f afrag[3];
#pragma unroll
    for (int c = 0; c < 3; ++c) {
        const int j0 = 4 * c + 2 * half;
        afrag[c].x = (m < NB && j0 + 1 < NK) ? b[mclp * NK + j0 + 1] * ra0 : 0.0f;
        afrag[c].y = (m < NB && j0     < NK) ? b[mclp * NK + j0    ] * ra0 : 0.0f;
    }

    // ---- IIR feedback coeffs for this lane's band ----
    const int   band = (lane < NB) ? lane : (NB - 1);
    const float rb0  = 1.0f / a[band * NK];
    float arec[NK - 1];
#pragma unroll
    for (int k = 1; k < NK; ++k) arec[k - 1] = a[band * NK + k] * rb0;

    float yh[NK - 1];
#pragma unroll
    for (int k = 0; k < NK - 1; ++k) yh[k] = 0.0f;

    __syncthreads();

    float* outp = out + ((size_t)seq * NB + band) * T_LEN;
    const int n = lane & 15;

    for (int tile = 0; tile < TILES; ++tile) {
        const int t0 = tile * 16;

        // ---- FIR tile: fir(16x16) = Bcoef(16x4) x Hankel(4x16), 3 K-chunks ----
        // bf = (x[t-(j0+1)], x[t-j0]) : ascending LDS addresses -> direct pair load.
        v8f d = {0.f, 0.f, 0.f, 0.f, 0.f, 0.f, 0.f, 0.f};
#pragma unroll
        for (int c = 0; c < 3; ++c) {
            const int base = 16 + t0 + n - (4 * c + 2 * half);  // >= 6 always
            v2f bf;
            bf.x = xw[base - 1];
            bf.y = xw[base];
            d = __builtin_amdgcn_wmma_f32_16x16x4_f32(
                    false, afrag[c], false, bf, (short)0, d, false, false);
        }

        // ---- scatter D tile (16 bands x 16 timesteps) to LDS ----
        // D layout: VGPR r, lane L -> M = r + 8*(L>>4), N = L&15
#pragma unroll
        for (int r = 0; r < 8; ++r)
            fw[(r + 8 * half) * FSTRIDE + n] = d[r];

        asm volatile("s_wait_dscnt 0" ::: "memory");  // cross-lane LDS handoff

        // ---- per-band recursion over this 16-step tile ----
        float fv[16];
        *(v4f*)&fv[0]  = *(const v4f*)&fw[band * FSTRIDE + 0];
        *(v4f*)&fv[4]  = *(const v4f*)&fw[band * FSTRIDE + 4];
        *(v4f*)&fv[8]  = *(const v4f*)&fw[band * FSTRIDE + 8];
        *(v4f*)&fv[12] = *(const v4f*)&fw[band * FSTRIDE + 12];

        float ybuf[16];
#pragma unroll
        for (int t = 0; t < 16; ++t) {
            float y = fv[t];
            // Oldest tap first: the loop-carried term arec[0]*y[t-1] is applied
            // LAST, so the serial dependence is ~1 FMA/step instead of 8.
#pragma unroll
            for (int k = NK - 2; k >= 0; --k) y -= arec[k] * yh[k];
#pragma unroll
            for (int k = NK - 2; k > 0; --k) yh[k] = yh[k - 1];
            yh[0]   = y;
            ybuf[t] = y;
        }

        if (lane < NB) {   // 64B contiguous per lane; NT hint: write-once stream
            __builtin_nontemporal_store(*(v4f*)&ybuf[0],  (v4f*)&outp[t0 + 0]);
            __builtin_nontemporal_store(*(v4f*)&ybuf[4],  (v4f*)&outp[t0 + 4]);
            __builtin_nontemporal_store(*(v4f*)&ybuf[8],  (v4f*)&outp[t0 + 8]);
            __builtin_nontemporal_store(*(v4f*)&ybuf[12], (v4f*)&outp[t0 + 12]);
        }
        // No WAR guard needed: DS ops are in-order per wave and the fir-tile
        // store/load addresses alias, which already pins compiler ordering.
    }
}

extern "C" void kernel_launch(void* const* d_in, const int* in_sizes, int n_in,
                              void* d_out, int out_size, void* d_ws, size_t ws_size,
                              hipStream_t stream) {
    const float* x = (const float*)d_in[0];   // (128, 22, 2000) f32
    const float* a = (const float*)d_in[1];   // (9, 9) f32
    const float* b = (const float*)d_in[2];   // (9, 9) f32
    float* out = (float*)d_out;               // (128, 22, 9, 2000) f32

    const int nseq   = in_sizes[0] / T_LEN;   // 2816
    const int blocks = (nseq + WAVES_PER_BLOCK - 1) / WAVES_PER_BLOCK;
    FilterBank_wmma_kernel<<<blocks, 32 * WAVES_PER_BLOCK, 0, stream>>>(x, a, b, out, nseq);
}